// Luong_Gen_Attention_6545530159367
// MI455X (gfx1250) — compile-verified
//
#include <hip/hip_runtime.h>
#include <math.h>

#define B_  16
#define S_  4096
#define D_  2048
#define A_  2048
#define SCHUNKS 16

typedef __attribute__((ext_vector_type(2))) float v2f;
typedef __attribute__((ext_vector_type(4))) float v4f;
typedef __attribute__((ext_vector_type(8))) float v8f;

// ---------------------------------------------------------------------------
// Kernel 1: hid_proj[16,A] = hid[16,D] @ W[A,D]^T + bias   (fp32 WMMA 16x16x4)
// One wave per 16-wide N-tile of A. M = batch (exactly 16). K loops over D.
// A-frag (16x4 f32, ISA 7.12.2): lane L -> m=L%16, VGPR j -> K = 2*(L>>4)+j.
// B-frag (4x16): lane L -> n=L%16, VGPR j -> K = 2*(L>>4)+j (B[k',n]=W[n0+n,k+k']).
// C/D (16x16 f32): VGPR r -> m = r + 8*(L>>4), n = L%16.
// ---------------------------------------------------------------------------
__global__ __launch_bounds__(32)
void hidproj_wmma(const float* __restrict__ hid, const float* __restrict__ W,
                  const float* __restrict__ bias, float* __restrict__ hp) {
    const int n0   = blockIdx.x * 16;
    const int lane = threadIdx.x;
    const int half = lane >> 4;
    const int lm   = lane & 15;
    const float* arow = hid + (long)lm * D_;        // A row m = lm
    const float* brow = W + (long)(n0 + lm) * D_;   // B col n = lm (row of W)
    v8f acc = {};
    for (int k = 0; k < D_; k += 4) {
        v2f a = *(const v2f*)(arow + k + 2 * half);
        v2f b = *(const v2f*)(brow + k + 2 * half);
        acc = __builtin_amdgcn_wmma_f32_16x16x4_f32(false, a, false, b,
                                                    (short)0, acc, false, false);
    }
#pragma unroll
    for (int r = 0; r < 8; ++r) {
        const int m = r + 8 * half;
        hp[(long)m * A_ + n0 + lm] = acc[r] + bias[n0 + lm];
    }
}

// ---------------------------------------------------------------------------
// Kernel 2: v[16,D] = hid_proj[16,A] @ W[A,D]   (fp32 WMMA 16x16x4)
// B[k',n] = W[k+k', d0+n].
// ---------------------------------------------------------------------------
__global__ __launch_bounds__(32)
void vproj_wmma(const float* __restrict__ hp, const float* __restrict__ W,
                float* __restrict__ v) {
    const int d0   = blockIdx.x * 16;
    const int lane = threadIdx.x;
    const int half = lane >> 4;
    const int lm   = lane & 15;
    const float* arow = hp + (long)lm * A_;
    v8f acc = {};
    for (int k = 0; k < A_; k += 4) {
        v2f a = *(const v2f*)(arow + k + 2 * half);
        v2f b;
        b.x = W[(long)(k + 2 * half)     * D_ + d0 + lm];
        b.y = W[(long)(k + 2 * half + 1) * D_ + d0 + lm];
        acc = __builtin_amdgcn_wmma_f32_16x16x4_f32(false, a, false, b,
                                                    (short)0, acc, false, false);
    }
#pragma unroll
    for (int r = 0; r < 8; ++r) {
        const int m = r + 8 * half;
        v[(long)m * D_ + d0 + lm] = acc[r];
    }
}

// ---------------------------------------------------------------------------
// Kernel 3: c[b] = bias . hid_proj[b,:]
// ---------------------------------------------------------------------------
__global__ __launch_bounds__(256)
void compute_c(const float* __restrict__ hp, const float* __restrict__ bias,
               float* __restrict__ c) {
    __shared__ float red[256];
    const int b = blockIdx.x;
    float acc = 0.f;
    for (int a = threadIdx.x; a < A_; a += 256) acc += hp[(long)b * A_ + a] * bias[a];
    red[threadIdx.x] = acc; __syncthreads();
    for (int off = 128; off > 0; off >>= 1) {
        if (threadIdx.x < off) red[threadIdx.x] += red[threadIdx.x + off];
        __syncthreads();
    }
    if (threadIdx.x == 0) c[b] = red[0];
}

// ---------------------------------------------------------------------------
// Kernel 4: scores[b,s] = enc[b,s,:] . v[b,:] + c[b]    (HBM pass #1, 512 MiB)
// One wave per (b,s) row; 128-bit coalesced loads; wave32 shuffle reduction.
// ---------------------------------------------------------------------------
__global__ __launch_bounds__(256)
void scores_kernel(const float* __restrict__ enc, const float* __restrict__ v,
                   const float* __restrict__ c, float* __restrict__ scores) {
    const int  wave = threadIdx.x >> 5;
    const int  lane = threadIdx.x & 31;
    const long row  = (long)blockIdx.x * 8 + wave;     // row = b*S + s
    const int  b    = (int)(row / S_);
    const float* erow = enc + row * (long)D_;
    const float* vrow = v + (long)b * D_;
    float acc = 0.f;
#pragma unroll 4
    for (int i = 0; i < D_ / 128; ++i) {
        const int d = i * 128 + lane * 4;
        v4f e = *(const v4f*)(erow + d);
        v4f w = *(const v4f*)(vrow + d);
        acc += e.x * w.x + e.y * w.y + e.z * w.z + e.w * w.w;
    }
#pragma unroll
    for (int off = 16; off > 0; off >>= 1) acc += __shfl_xor(acc, off, 32);
    if (lane == 0) scores[row] = acc + c[b];
}

// ---------------------------------------------------------------------------
// Kernel 5: softmax over S per batch row; probs into workspace.
// ---------------------------------------------------------------------------
__global__ __launch_bounds__(256)
void softmax_kernel(const float* __restrict__ scores, float* __restrict__ probs) {
    __shared__ float red[256];
    const int b = blockIdx.x;
    const float* row = scores + (long)b * S_;
    float m = -INFINITY;
    for (int s = threadIdx.x; s < S_; s += 256) m = fmaxf(m, row[s]);
    red[threadIdx.x] = m; __syncthreads();
    for (int off = 128; off > 0; off >>= 1) {
        if (threadIdx.x < off) red[threadIdx.x] = fmaxf(red[threadIdx.x], red[threadIdx.x + off]);
        __syncthreads();
    }
    m = red[0]; __syncthreads();
    float sum = 0.f;
    for (int s = threadIdx.x; s < S_; s += 256) sum += __expf(row[s] - m);
    red[threadIdx.x] = sum; __syncthreads();
    for (int off = 128; off > 0; off >>= 1) {
        if (threadIdx.x < off) red[threadIdx.x] += red[threadIdx.x + off];
        __syncthreads();
    }
    const float inv = 1.f / red[0];
    for (int s = threadIdx.x; s < S_; s += 256) probs[(long)b * S_ + s] = __expf(row[s] - m) * inv;
}

// ---------------------------------------------------------------------------
// Kernel 6: context partials (HBM pass #2, 512 MiB). Deterministic: each block
// owns (b, 1024-wide d-chunk, 256-long s-chunk); partial sums to workspace.
// ---------------------------------------------------------------------------
__global__ __launch_bounds__(256)
void context_partial(const float* __restrict__ enc, const float* __restrict__ probs,
                     float* __restrict__ partial) {
    const int dchunk = blockIdx.x;          // 0..1
    const int sc     = blockIdx.y;          // 0..SCHUNKS-1
    const int b      = blockIdx.z;          // 0..15
    const int d      = dchunk * 1024 + threadIdx.x * 4;
    const int s0     = sc * (S_ / SCHUNKS);
    v4f acc = {};
#pragma unroll 4
    for (int s = s0; s < s0 + S_ / SCHUNKS; ++s) {
        const float p = probs[(long)b * S_ + s];
        v4f e = *(const v4f*)(enc + ((long)b * S_ + s) * D_ + d);
        acc.x += p * e.x; acc.y += p * e.y; acc.z += p * e.z; acc.w += p * e.w;
    }
    *(v4f*)(partial + ((long)(b * SCHUNKS + sc)) * D_ + d) = acc;
}

// ---------------------------------------------------------------------------
// Kernel 7: reduce partials -> context[b,d].
// ---------------------------------------------------------------------------
__global__ __launch_bounds__(256)
void context_reduce(const float* __restrict__ partial, float* __restrict__ ctx) {
    const int gid = blockIdx.x * 256 + threadIdx.x;    // float4 index
    const int b = (gid * 4) / D_;
    const int d = (gid * 4) % D_;
    v4f acc = {};
#pragma unroll
    for (int c2 = 0; c2 < SCHUNKS; ++c2) {
        v4f p = *(const v4f*)(partial + ((long)(b * SCHUNKS + c2)) * D_ + d);
        acc.x += p.x; acc.y += p.y; acc.z += p.z; acc.w += p.w;
    }
    *(v4f*)(ctx + (long)b * D_ + d) = acc;
}

extern "C" void kernel_launch(void* const* d_in, const int* in_sizes, int n_in,
                              void* d_out, int out_size, void* d_ws, size_t ws_size,
                              hipStream_t stream) {
    const float* enc  = (const float*)d_in[0];   // [B,S,D]
    const float* hid  = (const float*)d_in[1];   // [B,D]
    const float* W    = (const float*)d_in[2];   // [A,D]
    const float* bias = (const float*)d_in[3];   // [A]

    float* scores = (float*)d_out;               // [B*S]
    float* ctx    = (float*)d_out + (long)B_ * S_; // [B*D]

    char* p = (char*)d_ws;
    float* hp      = (float*)p; p += (long)B_ * A_ * sizeof(float);   // 128 KB
    float* v       = (float*)p; p += (long)B_ * D_ * sizeof(float);   // 128 KB
    float* c       = (float*)p; p += 64;                              // B floats
    float* probs   = (float*)p; p += (long)B_ * S_ * sizeof(float);   // 256 KB
    float* partial = (float*)p;                                       // 2 MB

    hidproj_wmma<<<A_ / 16, 32, 0, stream>>>(hid, W, bias, hp);
    vproj_wmma  <<<D_ / 16, 32, 0, stream>>>(hp, W, v);
    compute_c   <<<B_, 256, 0, stream>>>(hp, bias, c);
    scores_kernel<<<(B_ * S_) / 8, 256, 0, stream>>>(enc, v, c, scores);
    softmax_kernel<<<B_, 256, 0, stream>>>(scores, probs);
    dim3 gctx(D_ / 1024, SCHUNKS, B_);
    context_partial<<<gctx, 256, 0, stream>>>(enc, probs, partial);
    context_reduce<<<(B_ * D_ / 4) / 256, 256, 0, stream>>>(partial, ctx);
}